// ScaleGNN_5944234738283
// MI455X (gfx1250) — compile-verified
//
#include <hip/hip_runtime.h>
#include <math.h>

// ScaleGNN forward for MI455X (gfx1250, wave32).
// fp32 WMMA (V_WMMA_F32_16X16X4_F32) for all GEMM-shaped work:
//   sim = xn@xn^T (17.2 GF), 3x masked-hop GEMMs (51.5 GF), x@W_in, H@W_out.
// sim (64MB) is L2-resident on MI455X (192MB L2); adj/rand (384MB) stream once.

typedef __attribute__((ext_vector_type(2))) float v2f;
typedef __attribute__((ext_vector_type(8))) float v8f;

static constexpr int NN = 4096;   // nodes
static constexpr int FF = 512;    // features (== channels)

__device__ __forceinline__ v8f wmma4(v2f a, v2f b, v8f c) {
    // D(16x16,f32) = A(16x4,f32) x B(4x16,f32) + C
    return __builtin_amdgcn_wmma_f32_16x16x4_f32(
        false, a, false, b, (short)0, c, false, false);
}

// ---------------- small prep kernels ----------------

__global__ void k_coef(const float* __restrict__ alpha,
                       const float* __restrict__ beta,
                       float* __restrict__ coef) {
    float a0 = alpha[0], a1 = alpha[1], a2 = alpha[2];
    float m = fmaxf(a0, fmaxf(a1, a2));
    float e0 = __expf(a0 - m), e1 = __expf(a1 - m), e2 = __expf(a2 - m);
    float s = e0 + e1 + e2;
    coef[0] = e0 / s; coef[1] = e1 / s; coef[2] = e2 / s;
    coef[3] = beta[0];
}

// inv[i] = 1 / max(||x_i||, 1e-12); one wave per row, 8 rows per block.
__global__ void __launch_bounds__(256) k_norms(const float* __restrict__ x,
                                               float* __restrict__ inv) {
    int wave = threadIdx.x >> 5, lane = threadIdx.x & 31;
    int row = blockIdx.x * 8 + wave;
    const float* xr = x + (size_t)row * FF;
    float s = 0.0f;
    for (int j = lane; j < FF; j += 32) { float v = xr[j]; s += v * v; }
    for (int off = 16; off > 0; off >>= 1) s += __shfl_down(s, off, 32);
    if (lane == 0) inv[row] = 1.0f / fmaxf(sqrtf(s), 1e-12f);
}

// ---------------- sim = xn @ xn^T via fp32 WMMA ----------------
// Block: 128 threads = 4 waves. Wave w: rows [i0+16w, +16), cols [j0, +64).
__global__ void __launch_bounds__(128) k_sim(const float* __restrict__ x,
                                             const float* __restrict__ inv,
                                             float* __restrict__ sim) {
    int wave = threadIdx.x >> 5, lane = threadIdx.x & 31;
    int half = lane >> 4, mr = lane & 15;
    int i0 = blockIdx.y * 64 + wave * 16;
    int j0 = blockIdx.x * 64;

    const float* aPtr = x + (size_t)(i0 + mr) * FF + 2 * half;
    const float* bPtr[4];
#pragma unroll
    for (int t = 0; t < 4; ++t)
        bPtr[t] = x + (size_t)(j0 + t * 16 + mr) * FF + 2 * half;

    v8f z = {0.f,0.f,0.f,0.f,0.f,0.f,0.f,0.f};
    v8f acc[4] = {z, z, z, z};

    for (int k = 0; k < FF; k += 4) {
        v2f a = *(const v2f*)(aPtr + k);
#pragma unroll
        for (int t = 0; t < 4; ++t) {
            v2f b = *(const v2f*)(bPtr[t] + k);
            acc[t] = wmma4(a, b, acc[t]);
        }
    }

    float invN[4];
#pragma unroll
    for (int t = 0; t < 4; ++t) invN[t] = inv[j0 + t * 16 + mr];
#pragma unroll
    for (int v = 0; v < 8; ++v) {
        int row = i0 + v + 8 * half;
        float si = inv[row];
#pragma unroll
        for (int t = 0; t < 4; ++t)
            sim[(size_t)row * NN + j0 + t * 16 + mr] = acc[t][v] * si * invN[t];
    }
}

// ---------------- generic M x 512 x 512 GEMM: out = A@B + bias ----------------
__global__ void __launch_bounds__(128) k_gemm512(const float* __restrict__ A,
                                                 const float* __restrict__ B,
                                                 const float* __restrict__ bias,
                                                 float* __restrict__ out) {
    int wave = threadIdx.x >> 5, lane = threadIdx.x & 31;
    int half = lane >> 4, mr = lane & 15;
    int i0 = blockIdx.y * 64 + wave * 16;
    int c0 = blockIdx.x * 64;

    const float* aPtr = A + (size_t)(i0 + mr) * 512 + 2 * half;
    v8f z = {0.f,0.f,0.f,0.f,0.f,0.f,0.f,0.f};
    v8f acc[4] = {z, z, z, z};

    for (int k = 0; k < 512; k += 4) {
        v2f a = *(const v2f*)(aPtr + k);
        int kb = k + 2 * half;
#pragma unroll
        for (int t = 0; t < 4; ++t) {
            int col = c0 + t * 16 + mr;
            v2f b;
            b.x = B[(size_t)kb * 512 + col];
            b.y = B[(size_t)(kb + 1) * 512 + col];
            acc[t] = wmma4(a, b, acc[t]);
        }
    }
#pragma unroll
    for (int v = 0; v < 8; ++v) {
        int row = i0 + v + 8 * half;
#pragma unroll
        for (int t = 0; t < 4; ++t) {
            int col = c0 + t * 16 + mr;
            out[(size_t)row * 512 + col] = acc[t][v] + bias[col];
        }
    }
}

// ---------------- per-hop top-8 threshold per row ----------------
// One wave per row (8 rows / 256-thread block). thr[row] = 8th largest of
// a_k * adj[row][:] * sim[row][:].
__global__ void __launch_bounds__(256) k_topk(const float* __restrict__ adj,
                                              const float* __restrict__ sim,
                                              const float* __restrict__ coef,
                                              int hop,
                                              float* __restrict__ thr) {
    __shared__ float cand[8][256];
    int wave = threadIdx.x >> 5, lane = threadIdx.x & 31;
    int row = blockIdx.x * 8 + wave;
    float ak = coef[hop];
    const float* ar = adj + (size_t)row * NN;
    const float* sr = sim + (size_t)row * NN;

    float t[8];
#pragma unroll
    for (int i = 0; i < 8; ++i) t[i] = -__builtin_inff();

    for (int j = lane; j < NN; j += 32) {
        float v = ak * ar[j] * sr[j];
        if (v > t[7]) {
            t[7] = v;
#pragma unroll
            for (int i = 7; i > 0; --i) {
                if (t[i] > t[i - 1]) { float tmp = t[i]; t[i] = t[i - 1]; t[i - 1] = tmp; }
            }
        }
    }
#pragma unroll
    for (int i = 0; i < 8; ++i) cand[wave][lane * 8 + i] = t[i];
    __syncthreads();
    if (lane == 0) {
        float s[8];
#pragma unroll
        for (int i = 0; i < 8; ++i) s[i] = -__builtin_inff();
        for (int j = 0; j < 256; ++j) {
            float v = cand[wave][j];
            if (v > s[7]) {
                s[7] = v;
#pragma unroll
                for (int i = 7; i > 0; --i) {
                    if (s[i] > s[i - 1]) { float tmp = s[i]; s[i] = s[i - 1]; s[i - 1] = tmp; }
                }
            }
        }
        thr[row] = s[7];
    }
}

// ---------------- masked hop GEMM: H_high (+)= a_k * (A' @ x) ----------------
// A'_ij = Akw_ij * (Akw_ij >= thr_i || rand_ij < 0.5), Akw = a_k*adj*sim,
// built on the fly per 16x4 WMMA A-fragment. Wave = 16 rows x 128 cols.
__global__ void __launch_bounds__(128) k_hop(const float* __restrict__ adj,
                                             const float* __restrict__ rmask,
                                             const float* __restrict__ sim,
                                             const float* __restrict__ x,
                                             const float* __restrict__ thr,
                                             const float* __restrict__ coef,
                                             int hop,
                                             float* __restrict__ Hh,
                                             int accum) {
    int wave = threadIdx.x >> 5, lane = threadIdx.x & 31;
    int half = lane >> 4, mr = lane & 15;
    int i0 = blockIdx.y * 64 + wave * 16;
    int c0 = blockIdx.x * 128;
    int row = i0 + mr;

    float ak = coef[hop];
    float tr = thr[row];
    const float* ar = adj   + (size_t)row * NN + 2 * half;
    const float* sr = sim   + (size_t)row * NN + 2 * half;
    const float* rr = rmask + (size_t)row * NN + 2 * half;

    v8f z = {0.f,0.f,0.f,0.f,0.f,0.f,0.f,0.f};
    v8f acc[8] = {z, z, z, z, z, z, z, z};

    for (int k = 0; k < NN; k += 4) {
        v2f ad = *(const v2f*)(ar + k);
        v2f sm = *(const v2f*)(sr + k);
        v2f rm = *(const v2f*)(rr + k);
        float s0 = ak * ad.x * sm.x;
        float s1 = ak * ad.y * sm.y;
        v2f af;
        af.x = ((s0 >= tr) || (rm.x < 0.5f)) ? s0 : 0.0f;
        af.y = ((s1 >= tr) || (rm.y < 0.5f)) ? s1 : 0.0f;
        int kb = k + 2 * half;
#pragma unroll
        for (int t = 0; t < 8; ++t) {
            int col = c0 + t * 16 + mr;
            v2f b;
            b.x = x[(size_t)kb * FF + col];
            b.y = x[(size_t)(kb + 1) * FF + col];
            acc[t] = wmma4(af, b, acc[t]);
        }
    }
#pragma unroll
    for (int v = 0; v < 8; ++v) {
        int orow = i0 + v + 8 * half;
#pragma unroll
        for (int t = 0; t < 8; ++t) {
            int col = c0 + t * 16 + mr;
            size_t idx = (size_t)orow * FF + col;
            float val = ak * acc[t][v];
            Hh[idx] = accum ? (Hh[idx] + val) : val;
        }
    }
}

// ---------------- H = relu(b*H_low + (1-b)*H_high) ----------------
__global__ void k_relu(const float* __restrict__ Hl, const float* __restrict__ Hh,
                       const float* __restrict__ coef, float* __restrict__ H) {
    int i = blockIdx.x * blockDim.x + threadIdx.x;
    float b = coef[3];
    float v = b * Hl[i] + (1.0f - b) * Hh[i];
    H[i] = fmaxf(v, 0.0f);
}

// ---------------- in-place row log_softmax over 512 cols ----------------
__global__ void __launch_bounds__(256) k_lsm(float* __restrict__ out) {
    __shared__ float red[256];
    int row = blockIdx.x;
    float* r = out + (size_t)row * 512;
    float v0 = r[threadIdx.x], v1 = r[threadIdx.x + 256];

    red[threadIdx.x] = fmaxf(v0, v1);
    __syncthreads();
    for (int s = 128; s > 0; s >>= 1) {
        if (threadIdx.x < s) red[threadIdx.x] = fmaxf(red[threadIdx.x], red[threadIdx.x + s]);
        __syncthreads();
    }
    float m = red[0];
    __syncthreads();

    red[threadIdx.x] = __expf(v0 - m) + __expf(v1 - m);
    __syncthreads();
    for (int s = 128; s > 0; s >>= 1) {
        if (threadIdx.x < s) red[threadIdx.x] += red[threadIdx.x + s];
        __syncthreads();
    }
    float lse = logf(red[0]) + m;
    r[threadIdx.x]       = v0 - lse;
    r[threadIdx.x + 256] = v1 - lse;
}

// ---------------- launch ----------------
extern "C" void kernel_launch(void* const* d_in, const int* in_sizes, int n_in,
                              void* d_out, int out_size, void* d_ws, size_t ws_size,
                              hipStream_t stream) {
    (void)in_sizes; (void)n_in; (void)out_size; (void)ws_size;
    const float* x     = (const float*)d_in[0];
    const float* adj   = (const float*)d_in[1];
    const float* rmask = (const float*)d_in[2];
    const float* W_in  = (const float*)d_in[3];
    const float* b_in  = (const float*)d_in[4];
    const float* W_out = (const float*)d_in[5];
    const float* b_out = (const float*)d_in[6];
    const float* alpha = (const float*)d_in[7];
    const float* beta  = (const float*)d_in[8];
    float* out = (float*)d_out;

    float* ws    = (float*)d_ws;
    float* inv   = ws;                         // 4096
    float* coef  = ws + 4096;                  // 4: a0,a1,a2,beta
    float* thr   = ws + 4096 + 16;             // 4096 (reused per hop)
    float* sim   = ws + 8224;                  // 4096*4096
    float* Hlow  = sim  + (size_t)NN * NN;     // 4096*512
    float* Hhigh = Hlow + (size_t)NN * FF;     // 4096*512
    float* H     = Hhigh + (size_t)NN * FF;    // 4096*512

    k_coef <<<1, 1, 0, stream>>>(alpha, beta, coef);
    k_norms<<<NN / 8, 256, 0, stream>>>(x, inv);
    k_sim  <<<dim3(NN / 64, NN / 64), 128, 0, stream>>>(x, inv, sim);
    k_gemm512<<<dim3(FF / 64, NN / 64), 128, 0, stream>>>(x, W_in, b_in, Hlow);

    for (int h = 0; h < 3; ++h) {
        const float* adjk = adj   + (size_t)h * NN * NN;
        const float* rmk  = rmask + (size_t)h * NN * NN;
        k_topk<<<NN / 8, 256, 0, stream>>>(adjk, sim, coef, h, thr);
        k_hop <<<dim3(FF / 128, NN / 64), 128, 0, stream>>>(adjk, rmk, sim, x, thr,
                                                            coef, h, Hhigh, h > 0 ? 1 : 0);
    }

    k_relu<<<(NN * FF) / 256, 256, 0, stream>>>(Hlow, Hhigh, coef, H);
    k_gemm512<<<dim3(FF / 64, NN / 64), 128, 0, stream>>>(H, W_out, b_out, out);
    k_lsm <<<NN, 256, 0, stream>>>(out);
}